// BEVEstimator_85246510891626
// MI455X (gfx1250) — compile-verified
//
#include <hip/hip_runtime.h>
#include <stdint.h>

// ---- problem constants (fixed by the reference) ----
#define B_      8
#define D_      472
#define C_      472
#define FH_     56
#define FW_     100
#define NPIX    (FH_*FW_)        // 5600
#define NPIXALL (B_*NPIX)        // 44800
#define NX_     160
#define NY_     160
#define LPITCH  33               // LDS row pitch in dwords (32 data + 1 pad); gcd(33,64)=1
#define RPW     59               // channel rows per wave: 472/8

typedef unsigned int u32x4 __attribute__((ext_vector_type(4)));
typedef int          i32x4 __attribute__((ext_vector_type(4)));
typedef int          i32x8 __attribute__((ext_vector_type(8)));

// ------------------------------------------------------------------
// 1) zero the 386.6 MB output with vectorized stores
// ------------------------------------------------------------------
__global__ void __launch_bounds__(256) k_zero(float4* __restrict__ out, int n4) {
    int i = blockIdx.x * 256 + threadIdx.x;
    if (i < n4) out[i] = float4{0.f, 0.f, 0.f, 0.f};
}

// ------------------------------------------------------------------
// 2) per-batch combine = rot @ inv(K)  (adjugate 3x3 inverse)
// ------------------------------------------------------------------
__global__ void k_combine(const float* __restrict__ intrins,
                          const float* __restrict__ rot,
                          float* __restrict__ comb) {
    int b = threadIdx.x;
    if (b >= B_) return;
    const float* K = intrins + b * 9;
    const float* R = rot + b * 9;
    float a00=K[0],a01=K[1],a02=K[2],a10=K[3],a11=K[4],a12=K[5],a20=K[6],a21=K[7],a22=K[8];
    float det = a00*(a11*a22 - a12*a21) - a01*(a10*a22 - a12*a20) + a02*(a10*a21 - a11*a20);
    float id = 1.0f / det;
    float inv[9];
    inv[0] =  (a11*a22 - a12*a21)*id;
    inv[1] = -(a01*a22 - a02*a21)*id;
    inv[2] =  (a01*a12 - a02*a11)*id;
    inv[3] = -(a10*a22 - a12*a20)*id;
    inv[4] =  (a00*a22 - a02*a20)*id;
    inv[5] = -(a00*a12 - a02*a10)*id;
    inv[6] =  (a10*a21 - a11*a20)*id;
    inv[7] = -(a00*a21 - a01*a20)*id;
    inv[8] =  (a00*a11 - a01*a10)*id;
    float* Cb = comb + b * 9;
    #pragma unroll
    for (int i = 0; i < 3; ++i)
        #pragma unroll
        for (int j = 0; j < 3; ++j)
            Cb[i*3+j] = R[i*3+0]*inv[0*3+j] + R[i*3+1]*inv[1*3+j] + R[i*3+2]*inv[2*3+j];
}

// ------------------------------------------------------------------
// 3) per-pixel: strided argmax over D (coalesced across lanes),
//    frustum transform, voxel index (-1 if invalid)
// ------------------------------------------------------------------
__global__ void __launch_bounds__(256) k_vox(const float* __restrict__ dl,
                                             const float* __restrict__ comb,
                                             int* __restrict__ voxflat) {
    int p = blockIdx.x * 256 + threadIdx.x;
    if (p >= NPIXALL) return;
    int b   = p / NPIX;
    int rem = p - b * NPIX;
    int h   = rem / FW_;
    int w   = rem - h * FW_;

    // argmax over depth bins (first occurrence of max, like jnp.argmax)
    const float* base = dl + (size_t)b * D_ * NPIX + rem;
    float best = base[0];
    int   bi   = 0;
    for (int d = 1; d < D_; ++d) {
        float v = base[(size_t)d * NPIX];
        if (v > best) { best = v; bi = d; }
    }
    float dm = (float)bi * 0.125f + 1.0f;   // metric depth

    const float XSTEP = 1600.0f / 99.0f;    // linspace(0, OGW, 100)
    const float YSTEP = 896.0f  / 55.0f;    // linspace(0, OGH, 56)
    float u = (float)w * XSTEP;
    float v = (float)h * YSTEP;
    float x0 = u * dm, x1 = v * dm, x2 = dm;

    const float* Cb = comb + b * 9;
    float px = Cb[0]*x0 + Cb[1]*x1 + Cb[2]*x2;
    float py = Cb[3]*x0 + Cb[4]*x1 + Cb[5]*x2;
    float pz = Cb[6]*x0 + Cb[7]*x1 + Cb[8]*x2;

    bool inb = (px >  1.0f) && (px < 41.0f) &&
               (py > -20.0f) && (py < 20.0f) &&
               (pz > -10.0f) && (pz < 10.0f);
    int vx = (int)floorf((px -  1.0f) / 0.25f);
    int vy = (int)floorf((py + 20.0f) / 0.25f);
    int vz = (int)floorf((pz + 10.0f) / 20.0f);
    bool ing = (vx >= 0) && (vx < NX_) && (vy >= 0) && (vy < NY_) && (vz >= 0) && (vz < 1);

    voxflat[p] = (inb && ing) ? (b * (NX_*NY_) + vx * NY_ + vy) : -1;
}

// ------------------------------------------------------------------
// 4) scatter: stage a 472ch x 32px feature tile into LDS (layout
//    [c][LPITCH] dwords), then coalesced f32 atomic adds with lanes
//    over channels. Staging uses the Tensor Data Mover when the
//    builtin exists (1 DMA per wave, TENSORcnt), else falls back to
//    per-lane async global->LDS B32 copies (ASYNCcnt).
// ------------------------------------------------------------------
__global__ void __launch_bounds__(256) k_scatter(const float* __restrict__ feat,
                                                 const int* __restrict__ voxflat,
                                                 float* __restrict__ out) {
    __shared__ float smem[C_ * LPITCH];   // 472*33*4 = 62,304 B
    __shared__ int   sflat[32];

    const int p0   = blockIdx.x * 32;     // NPIX%32==0 -> tile never straddles a batch
    const int b    = p0 / NPIX;
    const int off0 = p0 - b * NPIX;

    if (threadIdx.x < 32) sflat[threadIdx.x] = voxflat[p0 + threadIdx.x];
    __syncthreads();

    // early-exit tiles with no valid pixel (uniform across block)
    int myf = sflat[threadIdx.x & 31];
    if (__ballot(myf >= 0) == 0ull) return;

    const int wave = threadIdx.x >> 5;
    const int lane = threadIdx.x & 31;
    const float* fb = feat + (size_t)b * C_ * NPIX + off0;

#if __has_builtin(__builtin_amdgcn_tensor_load_to_lds)
    // ---- TDM path: each wave DMAs its 59-row x 32-elem strided chunk ----
    {
        const int c0 = wave * RPW;
        unsigned long long ga =
            (unsigned long long)(uintptr_t)(fb + (size_t)c0 * NPIX);
        unsigned lds0 = (unsigned)(uintptr_t)(&smem[c0 * LPITCH]);
        // descriptor dwords must be wave-uniform SGPR values
        unsigned ga_lo = (unsigned)__builtin_amdgcn_readfirstlane((int)(unsigned)ga);
        unsigned ga_hi = (unsigned)__builtin_amdgcn_readfirstlane((int)(unsigned)(ga >> 32));
        unsigned ldsu  = (unsigned)__builtin_amdgcn_readfirstlane((int)lds0);

        u32x4 g0;
        g0.x = 1u;                                        // count=1 valid, user mode
        g0.y = ldsu;                                      // lds_addr (bytes)
        g0.z = ga_lo;                                     // global_addr[31:0]
        g0.w = (ga_hi & 0x01FFFFFFu) | 0x80000000u;       // global_addr[56:32] | type=2

        i32x8 g1;
        g1[0] = (2 << 16)      // data_size = 4 bytes
              | (1 << 20)      // pad_enable
              | (4 << 22)      // pad_interval: 32 dwords (one 128B row)
              | (0 << 25);     // pad_amount: 1 dword  -> LDS pitch 33 dwords
        g1[1] = (NPIX & 0xFFFF) << 16;                    // tensor_dim0 lo16 (=5600), barrier addr 0
        g1[2] = ((NPIX >> 16) & 0xFFFF)                   // tensor_dim0 hi16
              | ((C_ & 0xFFFF) << 16);                    // tensor_dim1 lo16 (=472)
        g1[3] = ((C_ >> 16) & 0xFFFF) | (32 << 16);       // tensor_dim1 hi16, tile_dim0 = 32
        g1[4] = RPW;                                      // tile_dim1 = 59, tile_dim2 = 0
        g1[5] = NPIX;                                     // tensor_dim0_stride lo32 = 5600
        g1[6] = 0;                                        // stride hi16, dim1_stride lo16
        g1[7] = 0;                                        // dim1_stride hi32

        i32x4 z4 = {0, 0, 0, 0};
#if defined(__clang_major__) && (__clang_major__ >= 23)
        i32x8 z8 = {0, 0, 0, 0, 0, 0, 0, 0};
        __builtin_amdgcn_tensor_load_to_lds(g0, g1, z4, z4, z8, 0);
#else
        __builtin_amdgcn_tensor_load_to_lds(g0, g1, z4, z4, 0);
#endif
    }
    __builtin_amdgcn_s_wait_tensorcnt(0);
#else
    // ---- fallback: per-lane async copy with transposed LDS addresses ----
    for (int idx = threadIdx.x; idx < C_ * 32; idx += 256) {   // 59 exact iters
        int c   = idx >> 5;
        int pix = idx & 31;
        unsigned long long gp = (unsigned long long)(uintptr_t)(fb + (size_t)c * NPIX + pix);
        unsigned lds = (unsigned)(uintptr_t)(&smem[c * LPITCH + pix]);
        asm volatile("global_load_async_to_lds_b32 %0, %1, off"
                     :: "v"(lds), "v"(gp) : "memory");
    }
    asm volatile("s_wait_asynccnt 0" ::: "memory");
#endif
    __syncthreads();

    // 8 waves x 4 pixels; lanes over channels -> consecutive atomic
    // addresses; LDS reads stride LPITCH=33 dwords -> conflict-free
    #pragma unroll
    for (int pp = 0; pp < 4; ++pp) {
        int pix  = wave * 4 + pp;
        int flat = sflat[pix];               // uniform within wave -> scalar branch
        if (flat < 0) continue;
        float* ob = out + (size_t)flat * C_;
        for (int c = lane; c < C_; c += 32) {
            __hip_atomic_fetch_add(&ob[c], smem[c * LPITCH + pix],
                                   __ATOMIC_RELAXED, __HIP_MEMORY_SCOPE_AGENT);
        }
    }
}

// ------------------------------------------------------------------
extern "C" void kernel_launch(void* const* d_in, const int* in_sizes, int n_in,
                              void* d_out, int out_size, void* d_ws, size_t ws_size,
                              hipStream_t stream) {
    const float* depth_logits = (const float*)d_in[0];   // [8,472,56,100]
    const float* features     = (const float*)d_in[1];   // [8,472,56,100]
    const float* intrins      = (const float*)d_in[2];   // [8,3,3]
    const float* rotMtx       = (const float*)d_in[3];   // [8,3,3]
    float* out = (float*)d_out;                          // [8,160,160,472]

    // workspace layout: [0,288) combine matrices, [1024, 1024+44800*4) voxel ids
    float* comb    = (float*)d_ws;
    int*   voxflat = (int*)((char*)d_ws + 1024);

    int n4 = out_size / 4;                               // 96,665,600 / 4, exact
    k_zero   <<<(n4 + 255) / 256, 256, 0, stream>>>((float4*)out, n4);
    k_combine<<<1, 32, 0, stream>>>(intrins, rotMtx, comb);
    k_vox    <<<NPIXALL / 256, 256, 0, stream>>>(depth_logits, comb, voxflat);
    k_scatter<<<NPIXALL / 32, 256, 0, stream>>>(features, voxflat, out);
}